// MahalanobisLoss2D_69045894250983
// MI455X (gfx1250) — compile-verified
//
#include <hip/hip_runtime.h>

typedef float v2f __attribute__((ext_vector_type(2)));
typedef float v8f __attribute__((ext_vector_type(8)));

#define DD 64            // series length
#define TILE_ROWS 16
#define LDS_STRIDE 68    // floats; 68*4=272 B = 17*16 -> float4-aligned rows, conflict-free col reads
#define WAVES 8
#define BLOCK (WAVES * 32)

__device__ __forceinline__ float vsel8(v8f v, int i) {
    float r = v[0];
    r = (i == 1) ? v[1] : r;
    r = (i == 2) ? v[2] : r;
    r = (i == 3) ? v[3] : r;
    r = (i == 4) ? v[4] : r;
    r = (i == 5) ? v[5] : r;
    r = (i == 6) ? v[6] : r;
    r = (i == 7) ? v[7] : r;
    return r;
}

__global__ void __launch_bounds__(BLOCK)
mahal_tile_kernel(const float* __restrict__ yt, const float* __restrict__ yp,
                  const float* __restrict__ param, const int* __restrict__ nrow,
                  float* __restrict__ partial, int numTiles)
{
    __shared__ float lds[WAVES][TILE_ROWS * LDS_STRIDE];
    __shared__ float rowacc[WAVES][TILE_ROWS];

    const int lane = threadIdx.x & 31;
    const int wave = threadIdx.x >> 5;

    // coef = -p/(1+p^2), p = 2*sigmoid(param)-1
    const float t = param[0];
    const float p = 2.0f / (1.0f + __expf(-t)) - 1.0f;
    const float coef = -p / (1.0f + p * p);

    const int m  = lane & 15;          // row of tile this lane represents in A / N col in C
    const int hi = lane >> 4;          // half-wave
    const int o  = hi ? 2 : 0;         // K-pair offset per f32 WMMA A/B layout
    const bool isDiag = ((m >> 3) == hi);
    const int dreg = m & 7;

    float acc = 0.0f;

    const int slots = gridDim.x * WAVES;
    const int iters = (numTiles + slots - 1) / slots;

    for (int it = 0; it < iters; ++it) {
        const int tile = (it * gridDim.x + blockIdx.x) * WAVES + wave;  // wave-uniform
        if (tile < numTiles) {
            const size_t base = (size_t)tile * (TILE_ROWS * DD);
            const int rowBase = tile * TILE_ROWS;

            // ---- stage masked diff tile (16x64) into this wave's LDS slab, coalesced float4 ----
            #pragma unroll
            for (int j = 0; j < 8; ++j) {
                const int idx = j * 32 + lane;       // float4 index in tile [0,256)
                const int r  = idx >> 4;             // tile row
                const int cv = idx & 15;             // float4 column
                const int c0 = cv << 2;
                const float4 a = *(const float4*)(yt + base + (size_t)r * DD + c0);
                const float4 b = *(const float4*)(yp + base + (size_t)r * DD + c0);
                const int nv = nrow[rowBase + r];
                float4 d;
                d.x = (c0 + 0 < nv) ? (a.x - b.x) : 0.0f;
                d.y = (c0 + 1 < nv) ? (a.y - b.y) : 0.0f;
                d.z = (c0 + 2 < nv) ? (a.z - b.z) : 0.0f;
                d.w = (c0 + 3 < nv) ? (a.w - b.w) : 0.0f;
                *(float4*)&lds[wave][r * LDS_STRIDE + c0] = d;
            }
            // wave-local LDS RAW: same-wave DS ops are in-order; wait for writes to land
            asm volatile("s_wait_dscnt 0" ::: "memory");

            // ---- two Gram diagonals via chained V_WMMA_F32_16X16X4_F32 ----
            const float* row = &lds[wave][m * LDS_STRIDE];
            v8f cg = {};   // diag -> sum d_i^2
            v8f cs = {};   // diag -> sum d_i * d_{i+1}
            #pragma unroll
            for (int k = 0; k < 16; ++k) {
                const int c = (k << 2) + o;
                const float a0 = row[c];
                const float a1 = row[c + 1];
                const float a2 = (c + 2 < DD) ? row[c + 2] : 0.0f;  // only k==15, hi half
                v2f A; A.x = a0; A.y = a1;      // A[m][c], A[m][c+1]
                v2f S; S.x = a1; S.y = a2;      // shifted-by-1 B column
                cg = __builtin_amdgcn_wmma_f32_16x16x4_f32(
                        false, A, false, A, (short)0, cg, false, false);
                cs = __builtin_amdgcn_wmma_f32_16x16x4_f32(
                        false, A, false, S, (short)0, cs, false, false);
            }

            if (isDiag) {
                const float d1 = vsel8(cg, dreg);
                const float d2 = vsel8(cs, dreg);
                const float quad = d1 + 2.0f * coef * d2;
                acc += quad / (float)nrow[rowBase + m];
            }
        }
    }

    if (isDiag) rowacc[wave][m] = acc;   // every (wave,row) slot written exactly once
    __syncthreads();
    if (threadIdx.x == 0) {
        float s = 0.0f;
        for (int w = 0; w < WAVES; ++w)
            for (int i = 0; i < TILE_ROWS; ++i)
                s += rowacc[w][i];       // fixed order -> deterministic
        partial[blockIdx.x] = s;
    }
}

__global__ void __launch_bounds__(256)
reduce_kernel(const float* __restrict__ partial, float* __restrict__ out,
              int nPart, float invB)
{
    __shared__ float s[256];
    float a = 0.0f;
    for (int i = threadIdx.x; i < nPart; i += 256) a += partial[i];
    s[threadIdx.x] = a;
    __syncthreads();
    for (int off = 128; off > 0; off >>= 1) {
        if (threadIdx.x < off) s[threadIdx.x] += s[threadIdx.x + off];
        __syncthreads();
    }
    if (threadIdx.x == 0) out[0] = s[0] * invB;
}

extern "C" void kernel_launch(void* const* d_in, const int* in_sizes, int n_in,
                              void* d_out, int out_size, void* d_ws, size_t ws_size,
                              hipStream_t stream)
{
    const float* yt    = (const float*)d_in[0];
    const float* yp    = (const float*)d_in[1];
    const float* param = (const float*)d_in[2];
    const int*   nvec  = (const int*)d_in[3];
    float* out = (float*)d_out;

    const long rows = (long)in_sizes[0] / DD;        // B = 1048576
    const int numTiles = (int)(rows / TILE_ROWS);    // rows is a multiple of 16

    int blocks = 2048;                               // 16 K waves; partials = 8 KB
    const long maxPart = (long)(ws_size / sizeof(float));
    if (maxPart > 0 && blocks > maxPart) blocks = (int)maxPart;
    if (blocks > numTiles) blocks = numTiles > 0 ? numTiles : 1;
    float* partial = (float*)d_ws;

    mahal_tile_kernel<<<blocks, BLOCK, 0, stream>>>(yt, yp, param, nvec, partial, numTiles);
    reduce_kernel<<<1, 256, 0, stream>>>(partial, out, blocks, 1.0f / (float)rows);
}